// ETE_13881334300901
// MI455X (gfx1250) — compile-verified
//
#include <hip/hip_runtime.h>
#include <stdint.h>

// ---- problem constants (from the reference) ----
#define NUM_ENTITY 100000
#define NUM_TYPE   5000
#define DIM        128
#define BATCH      512
#define MARGIN     2.0f

// ---- tiling ----
#define BM        32      // entity rows per workgroup
#define BN        64      // type rows per workgroup (one TDM tile)
#define LDSW      132     // LDS row stride in floats: 128 + 4 pad (via TDM pad) -> conflict-free b128
#define NTHREADS  256

typedef unsigned int u32x4 __attribute__((ext_vector_type(4)));
typedef int          i32x4 __attribute__((ext_vector_type(4)));
typedef int          i32x8 __attribute__((ext_vector_type(8)));

#if __has_builtin(__builtin_amdgcn_tensor_load_to_lds)
#define HAVE_TDM 1
#else
#define HAVE_TDM 0
#endif

__device__ __forceinline__ uint32_t lds_lo32(const void* p) {
  // Generic pointer into LDS = {shared-aperture base | 32-bit LDS offset};
  // low 32 bits are the wave-relative LDS byte address the TDM D# wants.
  return (uint32_t)(uintptr_t)p;
}

#if HAVE_TDM
__device__ __forceinline__ void tdm_load_type_tile(const float* gsrc, uint32_t lds_byte,
                                                   unsigned rows_rem) {
  // ---- D# group 0 (128b) ----
  unsigned long long ga = (unsigned long long)(uintptr_t)gsrc;
  u32x4 g0;
  g0[0] = 1u;                                    // [1:0] count = 1 (valid descriptor)
  g0[1] = lds_byte;                              // [63:32] lds_addr (bytes)
  g0[2] = (unsigned)ga;                          // [95:64] global_addr[31:0] (tile start)
  g0[3] = ((unsigned)(ga >> 32) & 0x01FFFFFFu)   // [120:96] global_addr[56:32]
          | 0x80000000u;                         // [127:126] type = 2 ("image")
  // ---- D# group 1 (256b) ----
  i32x8 g1;
  g1[0] = (2 << 16)       // data_size = 4 bytes
        | (1 << 20)       // pad_enable
        | (6 << 22)       // pad_interval = 128 dwords (one 128-float row)
        | (3 << 25);      // pad_amount   = 4 dwords  -> LDS row stride = 132 floats
  g1[1] = (int)(((unsigned)DIM) << 16);                       // tensor_dim0 lo16 = 128
  g1[2] = (int)((rows_rem & 0xFFFFu) << 16);                  // dim0 hi16 = 0 | tensor_dim1 lo16
  g1[3] = (int)((rows_rem >> 16) | (((unsigned)DIM) << 16));  // tensor_dim1 hi16 | tile_dim0 = 128
  g1[4] = BN;                                                 // tile_dim1 = 64 rows; tile_dim2 = 0
  g1[5] = DIM;                                                // tensor_dim0_stride = 128 elements
  g1[6] = 0;                                                  // stride0 hi16 = 0 | stride1 lo16 = 0
  g1[7] = 0;                                                  // stride1 hi = 0
  // ---- groups 2/3: 2D tile, higher dims unused ----
  i32x4 gz = {0, 0, 0, 0};
#if defined(__clang_major__) && (__clang_major__ >= 23)
  i32x8 gz8 = {0, 0, 0, 0, 0, 0, 0, 0};
  __builtin_amdgcn_tensor_load_to_lds(g0, g1, gz, gz, gz8, 0);
#else
  __builtin_amdgcn_tensor_load_to_lds(g0, g1, gz, gz, 0);
#endif
}
#endif  // HAVE_TDM

__global__ __launch_bounds__(NTHREADS)
void ETE_l1sig_kernel(const float* __restrict__ ent,
                      const float* __restrict__ typ,
                      const int*   __restrict__ xb,
                      float* __restrict__ out) {
  __shared__ __align__(16) float sE[BM * LDSW];   // 16,896 B
  __shared__ __align__(16) float sT[BN * LDSW];   // 33,792 B (TDM writes rows at stride 132)

  const int tid = threadIdx.x;
  const int b0  = blockIdx.x * BM;   // entity-row tile origin
  const int t0  = blockIdx.y * BN;   // type-row tile origin

#if HAVE_TDM
  // Wave 0 kicks off the async TDM copy of the 64x128 f32 type tile into LDS.
  if (tid < 32) {
    tdm_load_type_tile(typ + (size_t)t0 * DIM, lds_lo32(sT), (unsigned)(NUM_TYPE - t0));
  }
#else
  // Fallback: cooperative load of the type tile (clamped rows; OOB outputs never written).
  {
    const int r = tid >> 2;          // 0..63
    const int q = tid & 3;           // 32-float chunk
    const int tr = (t0 + r < NUM_TYPE) ? (t0 + r) : (NUM_TYPE - 1);
    const float* src = typ + (size_t)tr * DIM + q * 32;
    float* dst = &sT[r * LDSW + q * 32];
#pragma unroll
    for (int f = 0; f < 8; ++f)
      *(float4*)(dst + f * 4) = *(const float4*)(src + f * 4);
  }
#endif

  // All threads: gather the entity tile (indexed rows) into LDS. Overlaps the TDM DMA.
  {
    const int r = tid >> 3;          // 0..31  (BM rows)
    const int q = tid & 7;           // 16-float chunk
    const int idx = xb[b0 + r];
    const float* src = ent + (size_t)idx * DIM + q * 16;
    float* dst = &sE[r * LDSW + q * 16];
#pragma unroll
    for (int f = 0; f < 4; ++f)
      *(float4*)(dst + f * 4) = *(const float4*)(src + f * 4);
  }

#if HAVE_TDM
  if (tid < 32) __builtin_amdgcn_s_wait_tensorcnt(0);
#endif
  __syncthreads();

  const int tx = tid & 15;   // -> type cols  j = tx + 16*ci, ci in [0,4)
  const int ty = tid >> 4;   // -> batch rows i = ty + 16*ri, ri in [0,2)

  float4 acc[2][4];
#pragma unroll
  for (int ri = 0; ri < 2; ++ri)
#pragma unroll
    for (int ci = 0; ci < 4; ++ci)
      acc[ri][ci] = make_float4(0.f, 0.f, 0.f, 0.f);

#pragma unroll 4
  for (int k = 0; k < DIM; k += 4) {
    float4 ev[2];
    float4 tv[4];
#pragma unroll
    for (int ri = 0; ri < 2; ++ri)
      ev[ri] = *(const float4*)&sE[(ty + 16 * ri) * LDSW + k];   // 2-addr broadcast per wave
#pragma unroll
    for (int ci = 0; ci < 4; ++ci)
      tv[ci] = *(const float4*)&sT[(tx + 16 * ci) * LDSW + k];   // stride 132 -> conflict-free
#pragma unroll
    for (int ri = 0; ri < 2; ++ri)
#pragma unroll
      for (int ci = 0; ci < 4; ++ci) {
        acc[ri][ci].x += fabsf(ev[ri].x - tv[ci].x);
        acc[ri][ci].y += fabsf(ev[ri].y - tv[ci].y);
        acc[ri][ci].z += fabsf(ev[ri].z - tv[ci].z);
        acc[ri][ci].w += fabsf(ev[ri].w - tv[ci].w);
      }
  }

#pragma unroll
  for (int ri = 0; ri < 2; ++ri) {
    const int b = b0 + ty + 16 * ri;
#pragma unroll
    for (int ci = 0; ci < 4; ++ci) {
      const int t = t0 + tx + 16 * ci;
      if (t < NUM_TYPE) {
        const float4 a = acc[ri][ci];
        const float d = (a.x + a.y) + (a.z + a.w);
        out[(size_t)b * NUM_TYPE + t] = 1.0f / (1.0f + __expf(d - MARGIN));
      }
    }
  }
}

extern "C" void kernel_launch(void* const* d_in, const int* in_sizes, int n_in,
                              void* d_out, int out_size, void* d_ws, size_t ws_size,
                              hipStream_t stream) {
  (void)in_sizes; (void)n_in; (void)d_ws; (void)ws_size; (void)out_size;
  const float* ent = (const float*)d_in[0];   // [100000,128] f32
  const float* typ = (const float*)d_in[1];   // [5000,128]  f32
  const int*   xb  = (const int*)d_in[2];     // [512] int32
  float*       out = (float*)d_out;           // [512,5000]  f32

  dim3 grid(BATCH / BM, (NUM_TYPE + BN - 1) / BN);  // 16 x 79
  ETE_l1sig_kernel<<<grid, NTHREADS, 0, stream>>>(ent, typ, xb, out);
}